// VoxelwiseSupConMSELoss_24300924960829
// MI455X (gfx1250) — compile-verified
//
#include <hip/hip_runtime.h>
#include <math.h>

typedef float v2f __attribute__((ext_vector_type(2)));
typedef float v8f __attribute__((ext_vector_type(8)));

#define TEMP   0.07f
#define EPSN   1e-12f
#define PPOS   4000
#define NPIX   8000
#define FEATN  128
#define NVOXN  32768
#define MSE_N  (128 * 32768)

// workspace layout (float offsets)
#define OFF_P    0
#define OFF_M    (OFF_P + NPIX * FEATN)
#define OFF_INV  (OFF_M + FEATN * FEATN)
#define OFF_DEN  (OFF_INV + NPIX)
#define OFF_SPOS (OFF_DEN + NPIX)
#define OFF_MSE  (OFF_SPOS + 1)

// dynamic LDS layout for the main kernel (float offsets)
#define LSTR  132                    // padded row stride: 528B, 16B-aligned, conflict-free
#define SPI   0                      // Pi tile: 64 x LSTR
#define SPJ   (SPI + 64 * LSTR)      // Pj tile: 64 x LSTR
#define SINVI (SPJ + 64 * LSTR)
#define SINVJ (SINVI + 64)
#define SROWI (SINVJ + 64)
#define SROWJ (SROWI + 64)
#define SSPOS (SROWJ + 64)
#define SMEMF (SSPOS + 1)

// ---------------- init: zero denom + scalar accumulators ----------------
__global__ void k_init(float* ws) {
    int i = blockIdx.x * 256 + threadIdx.x;
    if (i < NPIX + 2) ws[OFF_DEN + i] = 0.0f;
}

// ---------------- gather rows of Zs and L2-normalize -> P[8000][128] ----
__global__ void k_gather(const float* __restrict__ Zs,
                         const int* __restrict__ pos,
                         const int* __restrict__ neg,
                         float* __restrict__ P) {
    int row  = blockIdx.x * 8 + (threadIdx.x >> 5);
    int lane = threadIdx.x & 31;
    if (row >= NPIX) return;
    int v = (row < PPOS) ? pos[row] : neg[row - PPOS];
    float x[4];
    float ss = 0.0f;
    #pragma unroll
    for (int q = 0; q < 4; ++q) {
        int f = lane + q * 32;
        x[q] = Zs[f * NVOXN + v];
        ss += x[q] * x[q];
    }
    #pragma unroll
    for (int m = 16; m >= 1; m >>= 1) ss += __shfl_xor(ss, m, 32);
    float nrm = sqrtf(ss);
    float sc  = 1.0f / fmaxf(nrm, EPSN);
    #pragma unroll
    for (int q = 0; q < 4; ++q) {
        int f = lane + q * 32;
        P[row * FEATN + f] = x[q] * sc;
    }
}

// ---------------- M = P^T P  (128x128, K=8000) --------------------------
__global__ void k_gram128(const float* __restrict__ P, float* __restrict__ M) {
    int a = blockIdx.y * 16 + threadIdx.y;
    int b = blockIdx.x * 16 + threadIdx.x;
    float s = 0.0f;
    for (int r = 0; r < NPIX; ++r)
        s = fmaf(P[r * FEATN + a], P[r * FEATN + b], s);
    M[a * FEATN + b] = s;
}

// ---------------- inv[i] = (1/T) / max(sqrt(p_i^T M p_i)/T, eps) --------
__global__ void k_inv(const float* __restrict__ P, const float* __restrict__ M,
                      float* __restrict__ inv) {
    __shared__ float sp[8][FEATN];
    int w = threadIdx.x >> 5, lane = threadIdx.x & 31;
    int row = blockIdx.x * 8 + w;
    #pragma unroll
    for (int q = 0; q < 4; ++q)
        sp[w][lane + q * 32] = P[row * FEATN + lane + q * 32];
    __syncthreads();
    float acc = 0.0f;
    #pragma unroll
    for (int q = 0; q < 4; ++q) {
        int f = lane * 4 + q;
        float mv = 0.0f;
        for (int g = 0; g < FEATN; ++g)
            mv = fmaf(M[f * FEATN + g], sp[w][g], mv);
        acc = fmaf(mv, sp[w][f], acc);
    }
    #pragma unroll
    for (int m = 16; m >= 1; m >>= 1) acc += __shfl_xor(acc, m, 32);
    if (lane == 0) {
        float rn = sqrtf(fmaxf(acc, 0.0f)) / TEMP;   // ||G_row||2 after /T
        inv[row] = (1.0f / TEMP) / fmaxf(rn, EPSN);
    }
}

// ---- main: triangular WMMA Gram tiles + fused two-sided supcon epilogue ----
// grid 125x125, upper triangle only (blockIdx.x >= blockIdx.y).
// Block tile 64(i) x 64(j); 4 waves, wave w owns rows [w*16, w*16+16), all 64 cols.
__global__ void __launch_bounds__(128)
k_gram_ep(const float* __restrict__ P, const float* __restrict__ inv,
          float* __restrict__ denom, float* __restrict__ spos_g) {
    if (blockIdx.x < blockIdx.y) return;       // symmetry: compute each pair once
    extern __shared__ float sm[];

    const int tid  = threadIdx.x;
    const int lane = tid & 31, w = tid >> 5;
    const int ib = blockIdx.y * 64;
    const int jb = blockIdx.x * 64;
    const bool diag = (blockIdx.x == blockIdx.y);

    // ---- stage Pi/Pj tiles via async global->LDS (ASYNCcnt), 16B chunks ----
    unsigned ldsPi = (unsigned)(size_t)(&sm[SPI]);
    unsigned ldsPj = (unsigned)(size_t)(&sm[SPJ]);
    for (int chunk = tid; chunk < 64 * 32; chunk += 128) {
        int r  = chunk >> 5;
        int c4 = (chunk & 31) << 2;
        unsigned di = ldsPi + (unsigned)((r * LSTR + c4) << 2);
        unsigned dj = ldsPj + (unsigned)((r * LSTR + c4) << 2);
        const float* gi = P + (size_t)(ib + r) * FEATN + c4;
        const float* gj = P + (size_t)(jb + r) * FEATN + c4;
        asm volatile("global_load_async_to_lds_b128 %0, %1, off"
                     :: "v"(di), "v"(gi) : "memory");
        asm volatile("global_load_async_to_lds_b128 %0, %1, off"
                     :: "v"(dj), "v"(gj) : "memory");
    }
    if (tid < 64) {
        sm[SINVI + tid] = inv[ib + tid];
        sm[SINVJ + tid] = inv[jb + tid];
        sm[SROWI + tid] = 0.0f;
        sm[SROWJ + tid] = 0.0f;
    }
    if (tid == 0) sm[SSPOS] = 0.0f;
    asm volatile("s_wait_asynccnt 0x0" ::: "memory");
    __syncthreads();

    const int half = lane >> 4;        // 0: lanes 0-15 (K=k0,k0+1)  1: lanes 16-31 (K=k0+2,k0+3)
    const int lh   = lane & 15;
    const float* Arow = &sm[SPI + (w * 16 + lh) * LSTR + half * 2];
    const float* Brow = &sm[SPJ + lh * LSTR + half * 2];

    v8f c[4] = {};
    #pragma unroll
    for (int k0 = 0; k0 < FEATN; k0 += 4) {
        v2f a = *(const v2f*)(Arow + k0);
        #pragma unroll
        for (int t = 0; t < 4; ++t) {
            v2f b = *(const v2f*)(Brow + t * 16 * LSTR + k0);
            c[t] = __builtin_amdgcn_wmma_f32_16x16x4_f32(false, a, false, b,
                                                         (short)0, c[t], false, false);
        }
    }

    // ---- epilogue: gn_ij = g*inv[i] (and mirrored gn_ji = g*inv[j]) --------
    float spos = 0.0f;
    float dI[8] = {0, 0, 0, 0, 0, 0, 0, 0};   // per-v diff-class exp sums -> denom[i]
    #pragma unroll
    for (int t = 0; t < 4; ++t) {
        const int  jcol = jb + t * 16 + lh;
        const bool jpos = jcol < PPOS;
        const float invj = sm[SINVJ + t * 16 + lh];
        float dJ = 0.0f;                      // per-lane diff-class exp sum -> denom[j]
        #pragma unroll
        for (int v = 0; v < 8; ++v) {
            int   mrow = w * 16 + v + half * 8;   // C layout: VGPR v, half -> M
            int   irow = ib + mrow;
            float g    = c[t][v];
            float gnij = g * sm[SINVI + mrow];
            bool  same = ((irow < PPOS) == jpos);
            if (same) {
                if (irow != jcol) spos += gnij;
            } else {
                dI[v] += __expf(gnij);
            }
            if (!diag) {                      // mirrored (j,i) element, i<j guaranteed
                float gnji = g * invj;
                if (same) spos += gnji;
                else      dJ  += __expf(gnji);
            }
        }
        if (!diag) atomicAdd(&sm[SROWJ + t * 16 + lh], dJ);   // ds_add_f32, no return
    }
    #pragma unroll
    for (int vv = 0; vv < 8; ++vv) {          // lane-rotated order: spread bank conflicts
        int v = (vv + lh) & 7;
        atomicAdd(&sm[SROWI + w * 16 + v + half * 8], dI[v]);
    }
    atomicAdd(&sm[SSPOS], spos);
    __syncthreads();

    if (tid < 64)          atomicAdd(&denom[ib + tid], sm[SROWI + tid]);
    if (!diag && tid < 64) atomicAdd(&denom[jb + tid], sm[SROWJ + tid]);
    if (tid == 0)          atomicAdd(spos_g, sm[SSPOS]);
}

// ---------------- MSE partial sums --------------------------------------
__global__ void k_mse(const float* __restrict__ a, const float* __restrict__ b,
                      float* __restrict__ msum) {
    __shared__ float red[8];
    float acc = 0.0f;
    for (int i = blockIdx.x * blockDim.x + threadIdx.x; i < MSE_N;
         i += gridDim.x * blockDim.x) {
        float d = a[i] - b[i];
        acc = fmaf(d, d, acc);
    }
    #pragma unroll
    for (int m = 16; m >= 1; m >>= 1) acc += __shfl_xor(acc, m, 32);
    int lane = threadIdx.x & 31, w = threadIdx.x >> 5;
    if (lane == 0) red[w] = acc;
    __syncthreads();
    if (threadIdx.x == 0) {
        float s = 0.0f;
        #pragma unroll
        for (int q = 0; q < 8; ++q) s += red[q];
        atomicAdd(msum, s);
    }
}

// ---------------- final combine ------------------------------------------
__global__ void k_final(const float* __restrict__ denom,
                        const float* __restrict__ spos,
                        const float* __restrict__ msum,
                        float* __restrict__ out) {
    __shared__ float red[8];
    float acc = 0.0f;
    for (int i = threadIdx.x; i < NPIX; i += 256) acc += __logf(denom[i]);
    #pragma unroll
    for (int m = 16; m >= 1; m >>= 1) acc += __shfl_xor(acc, m, 32);
    int lane = threadIdx.x & 31, w = threadIdx.x >> 5;
    if (lane == 0) red[w] = acc;
    __syncthreads();
    if (threadIdx.x == 0) {
        float logsum = 0.0f;
        #pragma unroll
        for (int q = 0; q < 8; ++q) logsum += red[q];
        float supcon = -(spos[0] / (float)NPIX) + logsum;
        float mse    = msum[0] / (float)MSE_N;
        out[0] = supcon + 0.5f * mse;
    }
}

extern "C" void kernel_launch(void* const* d_in, const int* in_sizes, int n_in,
                              void* d_out, int out_size, void* d_ws, size_t ws_size,
                              hipStream_t stream) {
    const float* Zs  = (const float*)d_in[0];
    const int*   pos = (const int*)d_in[1];
    const int*   neg = (const int*)d_in[2];
    const float* fd  = (const float*)d_in[3];
    const float* idf = (const float*)d_in[4];
    float* ws   = (float*)d_ws;
    float* P    = ws + OFF_P;
    float* M    = ws + OFF_M;
    float* inv  = ws + OFF_INV;
    float* den  = ws + OFF_DEN;
    float* spos = ws + OFF_SPOS;
    float* msum = ws + OFF_MSE;

    k_init   <<<(NPIX + 2 + 255) / 256, 256, 0, stream>>>(ws);
    k_gather <<<NPIX / 8, 256, 0, stream>>>(Zs, pos, neg, P);
    k_gram128<<<dim3(8, 8), dim3(16, 16), 0, stream>>>(P, M);
    k_inv    <<<NPIX / 8, 256, 0, stream>>>(P, M, inv);
    k_gram_ep<<<dim3(NPIX / 64, NPIX / 64), 128, SMEMF * sizeof(float), stream>>>
             (P, inv, den, spos);
    k_mse    <<<2048, 256, 0, stream>>>(fd, idf, msum);
    k_final  <<<1, 256, 0, stream>>>(den, spos, msum, (float*)d_out);
}